// EdgeEncodingPlus_76433237999807
// MI455X (gfx1250) — compile-verified
//
#include <hip/hip_runtime.h>

typedef __attribute__((ext_vector_type(2))) float v2f;
typedef __attribute__((ext_vector_type(4))) float v4f;
typedef __attribute__((ext_vector_type(8))) float v8f;
typedef __attribute__((ext_vector_type(4))) int   v4i;

#define PATH_LEN 8
#define EDGE_DIM 16

// ---------------------------------------------------------------------------
// Kernel 1: zero-fill the (nn x nn) output with non-temporal b128 stores
// (d_out is poisoned; reference output is zeros except scattered entries).
// NT keeps the 256 MB stream from evicting edge_attr out of L2.
// ---------------------------------------------------------------------------
__global__ __launch_bounds__(256) void ee_zero_kernel(float* __restrict__ out,
                                                      size_t n) {
  const size_t stride = (size_t)gridDim.x * blockDim.x * 4;
  size_t i = ((size_t)blockIdx.x * blockDim.x + threadIdx.x) * 4;
  const v4f z = {0.f, 0.f, 0.f, 0.f};
  for (; i + 3 < n; i += stride) {
    __builtin_nontemporal_store(z, reinterpret_cast<v4f*>(out + i));
  }
  if (blockIdx.x == 0 && threadIdx.x == 0) {
    for (size_t t = n & ~(size_t)3; t < n; ++t) out[t] = 0.f;
  }
}

// ---------------------------------------------------------------------------
// Kernel 2: one wave32 per 16 pairs.
//   dot[p] = sum_{l,d} mask(p,l) * edge_attr[paths[p][l]][d] * edge_vector[l][d]
// as 32 accumulating V_WMMA_F32_16X16X4_F32, B broadcast across columns.
//
// Chunk mapping (l, s): K0/K1 (lanes 0-15, VGPR0/1) cover d = s*2, s*2+1;
// K2/K3 (lanes 16-31) cover d = 8+s*2, 8+s*2+1.  So across s=0..3 each lane
// consumes the contiguous 32B slice EF[m,l][half*8 .. half*8+7]:
// two global_load_b128 per l, fully unconditional (mask folded in as *1/*0).
// ---------------------------------------------------------------------------
__global__ __launch_bounds__(256) void ee_main_kernel(
    const float* __restrict__ edge_attr,   // (E, 16)
    const int* __restrict__ src,           // (P,)
    const int* __restrict__ dst,           // (P,)
    const int* __restrict__ paths,         // (P, 8), -1 = padding
    const float* __restrict__ edge_vector, // (8, 16)
    float* __restrict__ out,               // (nn, nn)
    int P, int nn) {
  __shared__ float lds_ev[PATH_LEN * EDGE_DIM];  // 128 floats
  if (threadIdx.x < PATH_LEN * EDGE_DIM) {
    lds_ev[threadIdx.x] = edge_vector[threadIdx.x];
  }
  __syncthreads();

  const int lane = threadIdx.x & 31;
  const int half = lane >> 4;       // which K-pair of each chunk this lane feeds
  const int m    = lane & 15;       // A-matrix row (pair within tile)
  const int dOfs = half * 8;        // this lane's contiguous d-slice

  const long long wave     = ((long long)blockIdx.x * blockDim.x + threadIdx.x) >> 5;
  const long long pairBase = wave * 16;          // wave-uniform
  if (pairBase >= (long long)P) return;          // uniform exit, EXEC stays full

  const long long p       = pairBase + m;
  const bool      inRange = (p < (long long)P);
  const long long pSafe   = inRange ? p : (long long)(P - 1);

  // ---- path indices for row m: two NT b128 loads (row is 32B, aligned) ----
  const v4i* prow = reinterpret_cast<const v4i*>(paths + pSafe * PATH_LEN);
  v4i pi0 = __builtin_nontemporal_load(prow);
  v4i pi1 = __builtin_nontemporal_load(prow + 1);
  int pidx[PATH_LEN] = {pi0[0], pi0[1], pi0[2], pi0[3],
                        pi1[0], pi1[1], pi1[2], pi1[3]};
  if (!inRange) {
#pragma unroll
    for (int l = 0; l < PATH_LEN; ++l) pidx[l] = -1;
  }

  // ---- hoist ALL gathers: 16 unconditional global_load_b128, pipelined ----
  v4f qa[PATH_LEN][2];
#pragma unroll
  for (int l = 0; l < PATH_LEN; ++l) {
    const int idx = pidx[l];
    const float* eb =
        edge_attr + (long long)(idx < 0 ? 0 : idx) * EDGE_DIM + dOfs;
    qa[l][0] = *reinterpret_cast<const v4f*>(eb);
    qa[l][1] = *reinterpret_cast<const v4f*>(eb + 4);
  }
  // fold padding mask in with plain VALU (no EXEC manipulation)
#pragma unroll
  for (int l = 0; l < PATH_LEN; ++l) {
    const float mf = (pidx[l] < 0) ? 0.f : 1.f;
    qa[l][0] *= mf;
    qa[l][1] *= mf;
  }

  // ---- 32 accumulating f32 WMMAs; B from LDS (broadcast ds_load_b128) ----
  v8f acc = {0.f, 0.f, 0.f, 0.f, 0.f, 0.f, 0.f, 0.f};
#pragma unroll
  for (int l = 0; l < PATH_LEN; ++l) {
    const v4f b0 = *reinterpret_cast<const v4f*>(&lds_ev[l * EDGE_DIM + dOfs]);
    const v4f b1 = *reinterpret_cast<const v4f*>(&lds_ev[l * EDGE_DIM + dOfs + 4]);
    v2f a, b;
    a[0] = qa[l][0][0]; a[1] = qa[l][0][1]; b[0] = b0[0]; b[1] = b0[1];
    acc = __builtin_amdgcn_wmma_f32_16x16x4_f32(false, a, false, b, (short)0, acc, false, false);
    a[0] = qa[l][0][2]; a[1] = qa[l][0][3]; b[0] = b0[2]; b[1] = b0[3];
    acc = __builtin_amdgcn_wmma_f32_16x16x4_f32(false, a, false, b, (short)0, acc, false, false);
    a[0] = qa[l][1][0]; a[1] = qa[l][1][1]; b[0] = b1[0]; b[1] = b1[1];
    acc = __builtin_amdgcn_wmma_f32_16x16x4_f32(false, a, false, b, (short)0, acc, false, false);
    a[0] = qa[l][1][2]; a[1] = qa[l][1][3]; b[0] = b1[2]; b[1] = b1[3];
    acc = __builtin_amdgcn_wmma_f32_16x16x4_f32(false, a, false, b, (short)0, acc, false, false);
  }

  // ---- epilogue: D[r + 8*half, n] in acc[r], all n equal.
  // Lanes 0-7 write rows 0-7, lanes 16-23 write rows 8-15: one select chain +
  // one predicated NT scatter (single EXEC toggle).
  const int m8 = lane & 7;
  float v = acc[0];
  if (m8 == 1) v = acc[1];
  if (m8 == 2) v = acc[2];
  if (m8 == 3) v = acc[3];
  if (m8 == 4) v = acc[4];
  if (m8 == 5) v = acc[5];
  if (m8 == 6) v = acc[6];
  if (m8 == 7) v = acc[7];
  v *= (1.0f / PATH_LEN);  // mean over full path length (matches reference)

  const long long pw = pairBase + m8 + half * 8;
  if (((lane & 15) < 8) && (pw < (long long)P)) {
    const int s_ = __builtin_nontemporal_load(src + pw);
    const int d_ = __builtin_nontemporal_load(dst + pw);
    __builtin_nontemporal_store(v, out + (size_t)s_ * (size_t)nn + (size_t)d_);
  }
}

// ---------------------------------------------------------------------------
// Host launcher
// ---------------------------------------------------------------------------
extern "C" void kernel_launch(void* const* d_in, const int* in_sizes, int n_in,
                              void* d_out, int out_size, void* d_ws, size_t ws_size,
                              hipStream_t stream) {
  // setup_inputs order: num_nodes(int,1), edge_attr(f32), src(i32), dst(i32),
  //                     paths(i32), edge_vector(f32)
  const float* edge_attr   = (const float*)d_in[1];
  const int*   src         = (const int*)d_in[2];
  const int*   dst         = (const int*)d_in[3];
  const int*   paths       = (const int*)d_in[4];
  const float* edge_vector = (const float*)d_in[5];
  float* out = (float*)d_out;

  const int P = in_sizes[2];  // number of (src,dst) pairs

  // out is (nn x nn); recover nn from out_size (8192 for the reference shapes)
  int nn = 1;
  while ((long long)nn * nn < (long long)out_size) ++nn;

  // 1) zero the output (256 MB, NT b128 stores)
  ee_zero_kernel<<<8192, 256, 0, stream>>>(out, (size_t)out_size);

  // 2) gather + WMMA dot + scatter: 1 wave per 16 pairs, 8 waves per block
  const long long tiles  = ((long long)P + 15) / 16;
  const long long blocks = (tiles + 7) / 8;
  ee_main_kernel<<<(int)blocks, 256, 0, stream>>>(edge_attr, src, dst, paths,
                                                  edge_vector, out, P, nn);
}